// GaussianMask_1580547974542
// MI455X (gfx1250) — compile-verified
//
#include <hip/hip_runtime.h>

// GaussianMask: K[b,c,k,h,w] = exp(-(X[h+dr_k, w+dc_k] - X[h,w])^2 / (2*bw^2))
// 5x5 neighborhood minus center (24 taps), zero-padded borders.
// Store-bandwidth-bound (302 MB out vs 12.6 MB in): LDS halo tile + float4 NT stores.

static constexpr int HH   = 512;
static constexpr int WW   = 512;
static constexpr int TH   = 8;     // tile rows per block
static constexpr int TW   = 128;   // tile cols per block (32 lanes * 4 px)
static constexpr int HALO = 2;
static constexpr int LH   = TH + 2 * HALO;  // 12
static constexpr int LW   = TW + 2 * HALO;  // 132
static constexpr int NK   = 24;
static constexpr int NTHREADS = 256;

typedef float v4f __attribute__((ext_vector_type(4)));

// -log2(e) / (2 * bandwidth^2), bandwidth = 1.0
static constexpr float NEGC = -0.72134752044448170368f;

__global__ __launch_bounds__(NTHREADS)
void gaussian_mask_kernel(const float* __restrict__ X, float* __restrict__ K) {
    __shared__ float tile[LH * LW];

    const int bc  = blockIdx.z;               // fused (b,c) plane index
    const int h0  = blockIdx.y * TH;
    const int w0  = blockIdx.x * TW;
    const int tid = threadIdx.x;

    const size_t plane = (size_t)HH * WW;
    const float* Xp = X + (size_t)bc * plane;

    // ---- Cooperative halo-tile load into LDS (zero padding outside image) ----
    for (int i = tid; i < LH * LW; i += NTHREADS) {
        const int r  = i / LW;
        const int c  = i - r * LW;
        const int gh = h0 + r - HALO;
        const int gw = w0 + c - HALO;
        float v = 0.0f;
        if ((unsigned)gh < (unsigned)HH && (unsigned)gw < (unsigned)WW)
            v = Xp[gh * WW + gw];
        tile[i] = v;
    }
    __syncthreads();

    // ---- Each thread: 4 consecutive pixels of one row ----
    const int row = tid >> 5;          // 0..7  (wave handles one tile row)
    const int c0  = (tid & 31) << 2;   // 0..124 (lane -> 4-px group, coalesced)

    // 5x8 register window: rows row..row+4, LDS cols c0..c0+7
    float win[5][8];
    #pragma unroll
    for (int r = 0; r < 5; ++r) {
        #pragma unroll
        for (int j = 0; j < 8; ++j) {
            win[r][j] = tile[(row + r) * LW + c0 + j];
        }
    }

    const float xi0 = win[2][2];
    const float xi1 = win[2][3];
    const float xi2 = win[2][4];
    const float xi3 = win[2][5];

    const int gh = h0 + row;
    float* outp = K + (size_t)bc * NK * plane + (size_t)gh * WW + (w0 + c0);

    #pragma unroll
    for (int k = 0; k < NK; ++k) {
        const int idx = (k < 12) ? k : k + 1;  // skip center tap
        const int dr  = idx / 5;               // 0..4 (window row)
        const int dc  = idx % 5;               // 0..4 (window col base)

        const float a0 = win[dr][dc + 0] - xi0;
        const float a1 = win[dr][dc + 1] - xi1;
        const float a2 = win[dr][dc + 2] - xi2;
        const float a3 = win[dr][dc + 3] - xi3;

        v4f y;
        y.x = __builtin_amdgcn_exp2f(a0 * a0 * NEGC);  // v_exp_f32
        y.y = __builtin_amdgcn_exp2f(a1 * a1 * NEGC);
        y.z = __builtin_amdgcn_exp2f(a2 * a2 * NEGC);
        y.w = __builtin_amdgcn_exp2f(a3 * a3 * NEGC);

        // Streaming 16B store, non-temporal: output (302 MB) is write-once and
        // exceeds L2; keep the hot input resident instead.
        __builtin_nontemporal_store(y, (v4f*)(outp + (size_t)k * plane));
    }
}

extern "C" void kernel_launch(void* const* d_in, const int* in_sizes, int n_in,
                              void* d_out, int out_size, void* d_ws, size_t ws_size,
                              hipStream_t stream) {
    const float* X = (const float*)d_in[0];
    float* K = (float*)d_out;

    const int planes = in_sizes[0] / (HH * WW);  // B*C = 12
    dim3 grid(WW / TW, HH / TH, planes);
    dim3 block(NTHREADS, 1, 1);
    gaussian_mask_kernel<<<grid, block, 0, stream>>>(X, K);
}